// LinearSEM_712964571328
// MI455X (gfx1250) — compile-verified
//
#include <hip/hip_runtime.h>

typedef __attribute__((ext_vector_type(2))) float v2f;
typedef __attribute__((ext_vector_type(4))) float v4f;
typedef __attribute__((ext_vector_type(8))) float v8f;

#define NVARS 128
#define XS_STRIDE 132  // 128 + 4 pad: keeps 16B alignment, perfect LDS bank spread

#if defined(__has_builtin)
#if __has_builtin(__builtin_amdgcn_sched_group_barrier)
#define SCHED_DS_THEN_WMMA()                              \
  do {                                                    \
    __builtin_amdgcn_sched_group_barrier(0x100, 8, 0);    \
    __builtin_amdgcn_sched_group_barrier(0x008, 8, 0);    \
  } while (0)
#endif
#endif
#ifndef SCHED_DS_THEN_WMMA
#define SCHED_DS_THEN_WMMA() do {} while (0)
#endif

// ---------------------------------------------------------------------------
// Kernel 1: build M = (I - L)^-1 * diag(d) folded GEMM operand, pre-swizzled
// into WMMA B-fragment order.  T satisfies (I-L)T = I; thread j owns column j
// of T (the forward substitution only ever reads its own column -> no
// barriers during substitution).  Fragment order: slot
//   u = ((kstep*8 + n)*32 + lane)*2 + s
// holds B[K = kstep*4 + (lane<16?0:2) + s][N = n*16 + (lane&15)],
// where B[K][N] = T[N][K] * d[K].
// ---------------------------------------------------------------------------
__global__ __launch_bounds__(128) void sem_build_M(const float* __restrict__ weight,
                                                   const float* __restrict__ mask,
                                                   float* __restrict__ Msf) {
  __shared__ float T[NVARS * NVARS];
  __shared__ float dsh[NVARS];
  const int j = threadIdx.x;
  dsh[j] = weight[j * NVARS + j] * mask[j * NVARS + j];

  for (int i = 0; i < NVARS; ++i) {
    float s = (i == j) ? 1.0f : 0.0f;
    for (int k = 0; k < i; ++k) {
      const float wik = weight[i * NVARS + k] * mask[i * NVARS + k];
      s = fmaf(wik, T[k * NVARS + j], s);  // column-local read
    }
    T[i * NVARS + j] = s;
  }
  __syncthreads();

  for (int u = j; u < NVARS * NVARS; u += NVARS) {
    const int s     = u & 1;
    const int lane  = (u >> 1) & 31;
    const int frag  = u >> 6;
    const int n     = frag & 7;
    const int kstep = frag >> 3;
    const int K = kstep * 4 + ((lane < 16) ? 0 : 2) + s;
    const int N = n * 16 + (lane & 15);
    Msf[u] = T[N * NVARS + K] * dsh[K];
  }
}

// ---------------------------------------------------------------------------
// Kernel 2: z[b, :] = x[b, :] @ M via V_WMMA_F32_16X16X4_F32.
// Block = 256 threads (8 waves), 128 batch rows per block.
// Wave w computes rows [block*128 + 16w, +16) x all 128 output columns:
// 8 accumulator tiles, 32 K-steps of K=4 WMMA each.  Software-pipelined with
// sched_group_barrier enforcing [DS-read prefetch group][WMMA group] per
// K-step so steady-state waits are s_wait_dscnt>0, not 0.
// ---------------------------------------------------------------------------
__global__ __launch_bounds__(256) void sem_gemm(const float* __restrict__ x,
                                                const float* __restrict__ Msf,
                                                float* __restrict__ z) {
  extern __shared__ float smem[];
  float* Ms = smem;                  // 16384 floats, fragment order
  float* Xs = smem + NVARS * NVARS;  // 128 rows x XS_STRIDE floats

  const int tid  = threadIdx.x;
  const int lane = tid & 31;
  const int wave = tid >> 5;
  const long long block_row = (long long)blockIdx.x * 128;

  // Cooperative loads: M (hot, regular) and the x tile (streamed, NT).
  {
    const v4f* M4  = (const v4f*)Msf;
    v4f*       Ms4 = (v4f*)Ms;
#pragma unroll
    for (int it = 0; it < 16; ++it) {
      const int t = tid + it * 256;
      Ms4[t] = M4[t];
    }
    const v4f* X4 = (const v4f*)(x + block_row * NVARS);
#pragma unroll
    for (int it = 0; it < 16; ++it) {
      const int t   = tid + it * 256;
      const int row = t >> 5;   // 32 float4 per 128-float row
      const int c   = t & 31;
      v4f v = __builtin_nontemporal_load(X4 + t);
      *(v4f*)(Xs + row * XS_STRIDE + c * 4) = v;
    }
  }
  __syncthreads();

  const int khalf = lane >> 4;  // 0: K+0/K+1 half, 1: K+2/K+3 half
  const int l15   = lane & 15;
  const int arow  = wave * 16 + l15;          // A fragment: M index = lane&15
  const int arow2 = arow * (XS_STRIDE / 2);   // row base in float2 units

  v8f acc[8] = {};
  const v2f* Xs2 = (const v2f*)Xs;
  const v2f* Ms2 = (const v2f*)Ms;

  // Prologue: fragments for k = 0.
  // A 16x4 f32 layout (ISA 7.12.2): v0=A[m][kk], v1=A[m][kk+1], kk=k+khalf*2.
  v2f a_cur = Xs2[arow2 + khalf];
  v2f b_cur[8];
#pragma unroll
  for (int n = 0; n < 8; ++n) b_cur[n] = Ms2[n * 32 + lane];

#pragma unroll
  for (int k = 0; k < NVARS - 4; k += 4) {
    const int kn = k + 4;
    // Prefetch next K-step fragments (issue loads before the WMMA chain).
    v2f a_nxt = Xs2[arow2 + (kn >> 1) + khalf];
    v2f b_nxt[8];
#pragma unroll
    for (int n = 0; n < 8; ++n) b_nxt[n] = Ms2[((kn >> 2) * 8 + n) * 32 + lane];

    // Current K-step: 8 WMMAs sharing the A fragment.
#pragma unroll
    for (int n = 0; n < 8; ++n) {
      acc[n] = __builtin_amdgcn_wmma_f32_16x16x4_f32(
          /*neg_a=*/false, a_cur, /*neg_b=*/false, b_cur[n],
          /*c_mod=*/(short)0, acc[n], /*reuse_a=*/false, /*reuse_b=*/false);
    }
    a_cur = a_nxt;
#pragma unroll
    for (int n = 0; n < 8; ++n) b_cur[n] = b_nxt[n];

    // Pipeline shape for this K-step: DS-read prefetch group, then WMMA group.
    SCHED_DS_THEN_WMMA();
  }
  // Epilogue: last K-step (k = 124).
#pragma unroll
  for (int n = 0; n < 8; ++n) {
    acc[n] = __builtin_amdgcn_wmma_f32_16x16x4_f32(
        /*neg_a=*/false, a_cur, /*neg_b=*/false, b_cur[n],
        /*c_mod=*/(short)0, acc[n], /*reuse_a=*/false, /*reuse_b=*/false);
  }

  // D layout: VGPR r -> row M=r (lanes 0-15) / M=r+8 (lanes 16-31), N=lane&15
  const long long zbase = (block_row + wave * 16 + khalf * 8) * NVARS;
#pragma unroll
  for (int n = 0; n < 8; ++n) {
#pragma unroll
    for (int r = 0; r < 8; ++r) {
      __builtin_nontemporal_store(acc[n][r],
                                  z + zbase + (long long)r * NVARS + n * 16 + l15);
    }
  }
}

// ---------------------------------------------------------------------------
extern "C" void kernel_launch(void* const* d_in, const int* in_sizes, int n_in,
                              void* d_out, int out_size, void* d_ws, size_t ws_size,
                              hipStream_t stream) {
  const float* x      = (const float*)d_in[0];
  const float* weight = (const float*)d_in[1];
  const float* mask   = (const float*)d_in[2];
  float*       z      = (float*)d_out;
  float*       Msf    = (float*)d_ws;  // 64 KB fragment-ordered M

  sem_build_M<<<1, 128, 0, stream>>>(weight, mask, Msf);

  const int batch  = in_sizes[0] / NVARS;  // 524288
  const int blocks = batch / 128;          // 4096
  const size_t smem = (size_t)(NVARS * NVARS + 128 * XS_STRIDE) * sizeof(float);
  sem_gemm<<<blocks, 256, smem, stream>>>(x, Msf, z);
}